// OverISS_T_26079041421566
// MI455X (gfx1250) — compile-verified
//
#include <hip/hip_runtime.h>

// ---------------------------------------------------------------------------
// OverISS-T for MI455X (gfx1250), fp32 throughout (reference semantics).
// W/H are dead for the output and eliminated. Y lives in d_out
// ([Y.real | Y.imag] planes) and stays resident in the 192MB L2.
//
// Per iteration (3 launches):
//   K1: sumf[b,c,n] = sum_f |Y|^2 via V_WMMA_F32_16X16X4_F32 (ones-GEMM)
//   K2: g, 1/sqrt(g), wt[b,c,n]
//   K3: mega-kernel — per (b,f) block stages Y[b,:,f,:] + wt in LDS,
//       applies the 1/sqrt(g) scale, then runs all 4 ISS + 20 dereverb
//       sequential sub-steps locally (wave32 shuffle reductions), and
//       writes Y back once.
// ---------------------------------------------------------------------------

#define NB   2
#define NC   4
#define NF   512
#define NFR  1000
#define NTAPS 5
#define NDELAY 1
#define NITER 5
#define EPSV   0.001f
#define LAPEPS 1e-5f

#define PLANE (NB*NC*NF*NFR)   // 4,096,000 floats per (re|im) plane

typedef float v2f __attribute__((ext_vector_type(2)));
typedef float v8f __attribute__((ext_vector_type(8)));

// --------------------------- K0: Y0 = W0 X ---------------------------------
// W0 = max(eye, EPS): Y0[c] = 0.999*X[c] + 0.001 * sum_d X[d]
__global__ __launch_bounds__(256)
void k_init_y(const float* __restrict__ Xr, const float* __restrict__ Xi,
              float* __restrict__ Yr, float* __restrict__ Yi) {
  int i = blockIdx.x * 256 + threadIdx.x;          // over NB*NF*NFR
  if (i >= NB * NF * NFR) return;
  int b  = i / (NF * NFR);
  int fn = i - b * (NF * NFR);
  float xr[NC], xi[NC], sr = 0.f, si = 0.f;
#pragma unroll
  for (int c = 0; c < NC; ++c) {
    int idx = (b * NC + c) * (NF * NFR) + fn;
    xr[c] = Xr[idx]; xi[c] = Xi[idx];
    sr += xr[c]; si += xi[c];
  }
#pragma unroll
  for (int c = 0; c < NC; ++c) {
    int idx = (b * NC + c) * (NF * NFR) + fn;
    Yr[idx] = 0.999f * xr[c] + 0.001f * sr;
    Yi[idx] = 0.999f * xi[c] + 0.001f * si;
  }
}

// ---------------- K1: sumf[b,c,n] = sum_f |Y|^2 via f32 WMMA ----------------
// D(16x16) = ones(16x4) * B(4x16) + C accumulated over 128 chunks of 4 freqs.
// A = ones makes the sum invariant to B's exact K-striping; lanes 0-15 feed
// K rows {0,1}, lanes 16-31 rows {2,3}; C/D row M=0 (lanes 0-15, VGPR0)
// carries the per-frame totals. Tail frames: clamped address + zero mask so
// EXEC is all-1s at every WMMA.
__global__ __launch_bounds__(256)
void k_sumf_wmma(const float* __restrict__ Yr, const float* __restrict__ Yi,
                 float* __restrict__ sumf) {
  int bc   = blockIdx.y;                 // 0..7  = b*NC+c
  int wave = threadIdx.x >> 5;           // 0..7
  int lane = threadIdx.x & 31;
  int n0   = blockIdx.x * 128 + wave * 16;
  if (n0 >= NFR) return;                 // wave-uniform exit
  int  j     = lane & 15;
  int  khalf = lane >> 4;
  int  n     = n0 + j;
  bool nv    = (n < NFR);
  int  nn    = nv ? n : 0;
  float msk  = nv ? 1.f : 0.f;
  const float* yr = Yr + bc * (NF * NFR);
  const float* yi = Yi + bc * (NF * NFR);

  v8f acc   = {0.f,0.f,0.f,0.f,0.f,0.f,0.f,0.f};
  v2f aones = {1.0f, 1.0f};
  for (int f0 = 0; f0 < NF; f0 += 4) {
    int r0 = f0 + khalf * 2;
    int i0 = r0 * NFR + nn;
    float a0r = yr[i0],       a0i = yi[i0];
    float a1r = yr[i0 + NFR], a1i = yi[i0 + NFR];
    v2f bv = { msk * (a0r*a0r + a0i*a0i), msk * (a1r*a1r + a1i*a1i) };
    acc = __builtin_amdgcn_wmma_f32_16x16x4_f32(false, aones, false, bv,
                                                (short)0, acc, false, false);
  }
  if (khalf == 0 && nv) sumf[bc * NFR + n] = acc[0];
}

// -------- K2: g, 1/sqrt(g), wt[b,c,n] = g / max(2*sqrt(sumf),1e-5) ----------
__global__ __launch_bounds__(256)
void k_weights(const float* __restrict__ sumf, float* __restrict__ wt,
               float* __restrict__ gsinv) {
  int bc = blockIdx.x;
  __shared__ float sred[256];
  __shared__ float gsh;
  float acc = 0.f;
  for (int n = threadIdx.x; n < NFR; n += 256) acc += sumf[bc * NFR + n];
  sred[threadIdx.x] = acc; __syncthreads();
  for (int s = 128; s > 0; s >>= 1) {
    if (threadIdx.x < s) sred[threadIdx.x] += sred[threadIdx.x + s];
    __syncthreads();
  }
  if (threadIdx.x == 0) {
    float mean = sred[0] * (1.0f / ((float)NF * (float)NFR));
    float g    = fmaxf(mean, EPSV);
    gsinv[bc]  = 1.0f / sqrtf(g);      // sqrt(g) >= 0.0316 > EPS
    gsh = g;
  }
  __syncthreads();
  float g = gsh;
  for (int n = threadIdx.x; n < NFR; n += 256) {
    float d = 2.0f * sqrtf(sumf[bc * NFR + n]);
    wt[bc * NFR + n] = g / fmaxf(d, LAPEPS);
  }
}

// --------- K3: fused scale + 4 ISS + 20 dereverb steps per (b,f) ------------
__global__ __launch_bounds__(256)
void k_steps(float* __restrict__ Yr, float* __restrict__ Yi,
             const float* __restrict__ Xr, const float* __restrict__ Xi,
             const float* __restrict__ wt, const float* __restrict__ gsinv) {
  __shared__ float sYr[NC][NFR];     // 16 KB
  __shared__ float sYi[NC][NFR];     // 16 KB
  __shared__ float sW [NC][NFR];     // 16 KB
  __shared__ float sred[8 * 12];     // 8 waves x 12 partials

  int bf = blockIdx.x;               // b*NF + f
  int b  = bf / NF, f = bf - b * NF;
  int tid  = threadIdx.x;
  int lane = tid & 31, wave = tid >> 5;

  // ---- load Y (applying 1/sqrt(g)) and wt into LDS ----
#pragma unroll
  for (int c = 0; c < NC; ++c) {
    float s    = gsinv[b * NC + c];
    int gbase  = ((b * NC + c) * NF + f) * NFR;
    int wbase  = (b * NC + c) * NFR;
    for (int n = tid; n < NFR; n += 256) {
      sYr[c][n] = Yr[gbase + n] * s;
      sYi[c][n] = Yi[gbase + n] * s;
      sW [c][n] = wt[wbase + n];
    }
  }
  __syncthreads();

  // ------------------------- ISS source steps -------------------------
  for (int src = 0; src < NC; ++src) {
    float anr[NC] = {0,0,0,0}, ani[NC] = {0,0,0,0}, ad[NC] = {0,0,0,0};
    for (int n = tid; n < NFR; n += 256) {
      float ysr = sYr[src][n], ysi = sYi[src][n];
      float ms  = ysr * ysr + ysi * ysi;
#pragma unroll
      for (int c = 0; c < NC; ++c) {
        float w = sW[c][n];
        float yr_ = sYr[c][n], yi_ = sYi[c][n];
        anr[c] += w * (yr_ * ysr + yi_ * ysi);   // Y * conj(Ys)
        ani[c] += w * (yi_ * ysr - yr_ * ysi);
        ad[c]  += w * ms;
      }
    }
    // wave32 shuffle reduction
#pragma unroll
    for (int off = 16; off > 0; off >>= 1) {
#pragma unroll
      for (int c = 0; c < NC; ++c) {
        anr[c] += __shfl_down(anr[c], off);
        ani[c] += __shfl_down(ani[c], off);
        ad[c]  += __shfl_down(ad[c],  off);
      }
    }
    if (lane == 0) {
#pragma unroll
      for (int c = 0; c < NC; ++c) {
        sred[wave * 12 + c * 3 + 0] = anr[c];
        sred[wave * 12 + c * 3 + 1] = ani[c];
        sred[wave * 12 + c * 3 + 2] = ad[c];
      }
    }
    __syncthreads();
    // every thread computes v[c] redundantly from the 8 wave partials
    float vr[NC], vi[NC];
    const float invN = 1.0f / (float)NFR;
#pragma unroll
    for (int c = 0; c < NC; ++c) {
      float nr = 0.f, ni = 0.f, dn = 0.f;
#pragma unroll
      for (int w = 0; w < 8; ++w) {
        nr += sred[w * 12 + c * 3 + 0];
        ni += sred[w * 12 + c * 3 + 1];
        dn += sred[w * 12 + c * 3 + 2];
      }
      nr *= invN; ni *= invN; dn *= invN;
      if (c == src) { vr[c] = 1.0f - 1.0f / sqrtf(fmaxf(dn, EPSV)); vi[c] = 0.0f; }
      else { float dd = fmaxf(dn, EPSV); vr[c] = nr / dd; vi[c] = ni / dd; }
    }
    // rank-1 update in LDS: Y[c] -= v[c] * Ys  (Ys read into regs first)
    for (int n = tid; n < NFR; n += 256) {
      float ysr = sYr[src][n], ysi = sYi[src][n];
#pragma unroll
      for (int c = 0; c < NC; ++c) {
        sYr[c][n] -= vr[c] * ysr - vi[c] * ysi;
        sYi[c][n] -= vr[c] * ysi + vi[c] * ysr;
      }
    }
    __syncthreads();
  }

  // ------------------------- dereverb steps -------------------------
  for (int src = 0; src < NC; ++src) {
    int xbase = ((b * NC + src) * NF + f) * NFR;   // original (unscaled) X row
    for (int tap = 0; tap < NTAPS; ++tap) {
      int shift = tap - (NTAPS + NDELAY);          // in [-6,-2]
      float anr[NC] = {0,0,0,0}, ani[NC] = {0,0,0,0}, ad[NC] = {0,0,0,0};
      for (int n = tid; n < NFR; n += 256) {
        int np = n + shift;
        float xsr = 0.f, xsi = 0.f;
        if (np >= 0) { xsr = Xr[xbase + np]; xsi = Xi[xbase + np]; }
        float ms = xsr * xsr + xsi * xsi;
#pragma unroll
        for (int c = 0; c < NC; ++c) {
          float w = sW[c][n];
          float yr_ = sYr[c][n], yi_ = sYi[c][n];
          anr[c] += w * (yr_ * xsr + yi_ * xsi);   // Y * conj(Xst)
          ani[c] += w * (yi_ * xsr - yr_ * xsi);
          ad[c]  += w * ms;
        }
      }
#pragma unroll
      for (int off = 16; off > 0; off >>= 1) {
#pragma unroll
        for (int c = 0; c < NC; ++c) {
          anr[c] += __shfl_down(anr[c], off);
          ani[c] += __shfl_down(ani[c], off);
          ad[c]  += __shfl_down(ad[c],  off);
        }
      }
      if (lane == 0) {
#pragma unroll
        for (int c = 0; c < NC; ++c) {
          sred[wave * 12 + c * 3 + 0] = anr[c];
          sred[wave * 12 + c * 3 + 1] = ani[c];
          sred[wave * 12 + c * 3 + 2] = ad[c];
        }
      }
      __syncthreads();
      float vr[NC], vi[NC];
#pragma unroll
      for (int c = 0; c < NC; ++c) {
        float nr = 0.f, ni = 0.f, dn = 0.f;
#pragma unroll
        for (int w = 0; w < 8; ++w) {
          nr += sred[w * 12 + c * 3 + 0];
          ni += sred[w * 12 + c * 3 + 1];
          dn += sred[w * 12 + c * 3 + 2];
        }
        float dd = fmaxf(dn, EPSV);
        vr[c] = nr / dd; vi[c] = ni / dd;
      }
      for (int n = tid; n < NFR; n += 256) {
        int np = n + shift;
        float xsr = 0.f, xsi = 0.f;
        if (np >= 0) { xsr = Xr[xbase + np]; xsi = Xi[xbase + np]; }
#pragma unroll
        for (int c = 0; c < NC; ++c) {
          sYr[c][n] -= vr[c] * xsr - vi[c] * xsi;
          sYi[c][n] -= vr[c] * xsi + vi[c] * xsr;
        }
      }
      __syncthreads();
    }
  }

  // ---- write Y back ----
#pragma unroll
  for (int c = 0; c < NC; ++c) {
    int gbase = ((b * NC + c) * NF + f) * NFR;
    for (int n = tid; n < NFR; n += 256) {
      Yr[gbase + n] = sYr[c][n];
      Yi[gbase + n] = sYi[c][n];
    }
  }
}

// ---------------------------------------------------------------------------
extern "C" void kernel_launch(void* const* d_in, const int* in_sizes, int n_in,
                              void* d_out, int out_size, void* d_ws, size_t ws_size,
                              hipStream_t stream) {
  const float* Xr = (const float*)d_in[0];
  const float* Xi = (const float*)d_in[1];
  float* Yr = (float*)d_out;          // output layout: [Y.real | Y.imag]
  float* Yi = Yr + PLANE;

  float* ws   = (float*)d_ws;
  float* WT   = ws;                   // [NB*NC][NFR]  : 8000 (pad 8192)
  float* SUMF = ws + 8192;            // [NB*NC][NFR]  : 8000 (pad 8192)
  float* GS   = ws + 16384;           // [NB*NC]       : 8   (1/sqrt(g))

  k_init_y<<<(NB * NF * NFR + 255) / 256, 256, 0, stream>>>(Xr, Xi, Yr, Yi);

  for (int it = 0; it < NITER; ++it) {
    k_sumf_wmma<<<dim3(8, NB * NC), 256, 0, stream>>>(Yr, Yi, SUMF);
    k_weights<<<NB * NC, 256, 0, stream>>>(SUMF, WT, GS);
    k_steps<<<NB * NF, 256, 0, stream>>>(Yr, Yi, Xr, Xi, WT, GS);
  }
}